// PureMambaLayer_80221399155258
// MI455X (gfx1250) — compile-verified
//
#include <hip/hip_runtime.h>

// ---------------------------------------------------------------------------
// Mamba layer for MI455X (gfx1250), fp32 end-to-end.
// B=8, SEQ=4096, D_MODEL=1024, D_STATE=16, M = B*SEQ = 32768.
// Memory-bound (~0.65 GB min traffic -> ~30us @ 23.3 TB/s), so GEMMs stay in
// fp32 on V_WMMA_F32_16X16X4_F32. Weight tiles are staged into LDS by the
// Tensor Data Mover (double-buffered, s_wait_tensorcnt), A fragments are
// register double-buffered, epilogues are fused.
// ---------------------------------------------------------------------------

#define B_SZ     8
#define SEQ      4096
#define D_MODEL  1024
#define D_STATE  16
#define M_ROWS   (B_SZ * SEQ)          // 32768

typedef __attribute__((ext_vector_type(2))) float v2f;
typedef __attribute__((ext_vector_type(8))) float v8f;
typedef __attribute__((ext_vector_type(4))) unsigned int v4u;
typedef __attribute__((ext_vector_type(8))) int v8i;
typedef __attribute__((ext_vector_type(4))) int v4i;

__device__ __forceinline__ float silu_f(float v) {
    return v * __builtin_amdgcn_rcpf(1.0f + __expf(-v));
}

// LDS row stride in floats for staged W tiles: 16 payload + 2 pad.
// TDM applies the pad (pad_interval = 16 DWORDs, pad_amount = 2 DWORDs),
// which makes the 16-lane ds_load_b64 fragment reads bank-conflict free
// (lr*18 mod 64 hits 16 distinct banks).
#define BROW 18

// ---------------------------------------------------------------------------
// TDM: DMA a ROWS x 16-float tile of W (row stride LDW floats) into LDS.
// D# group0/group1 packed per CDNA5 ISA 8.3/8.4; groups 2/3 unused (2D tile).
// This toolchain's builtin takes 6 args (g0, g1, g2, g3, g4-unused, cpol).
// ---------------------------------------------------------------------------
template <int ROWS, int LDW>
__device__ __forceinline__ void tdm_load_tile(const float* gsrc, unsigned lds_byte_off)
{
    unsigned long long ga = (unsigned long long)(uintptr_t)gsrc;
    v4u g0;
    g0.x = 1u;                                                  // count=1
    g0.y = lds_byte_off;                                        // LDS dest byte addr
    g0.z = (unsigned)ga;                                        // global addr [31:0]
    g0.w = (unsigned)((ga >> 32) & 0x01FFFFFFu) | (2u << 30);   // addr [56:32] | type=2
    v8i g1;
    g1[0] = (2 << 16)          // data_size = 4 bytes
          | (1 << 20)          // pad_enable
          | (3 << 22)          // pad_interval: every 16 DWORDs
          | (1 << 25);         // pad_amount: 2 DWORDs -> LDS row stride 18 floats
    g1[1] = (LDW & 0xFFFF) << 16;                    // tensor_dim0[15:0]
    g1[2] = (LDW >> 16) | ((ROWS & 0xFFFF) << 16);   // tensor_dim0[31:16] | tensor_dim1[15:0]
    g1[3] = (16 << 16);                              // tensor_dim1[31:16] | tile_dim0 = 16
    g1[4] = ROWS;                                    // tile_dim1 = ROWS, tile_dim2 = 0
    g1[5] = LDW;                                     // tensor_dim0_stride[31:0]
    g1[6] = 0;
    g1[7] = 0;
    v4i z4 = {0, 0, 0, 0};
    v8i z8 = {0, 0, 0, 0, 0, 0, 0, 0};
    __builtin_amdgcn_tensor_load_to_lds(g0, g1, z4, z4, z8, 0);
}

// ---------------------------------------------------------------------------
// WMMA GEMM: out[m][n] = epi( sum_k A[m][k]*W[n][k] + bias[n] )
// Block: 256 threads = 8 waves; block tile 128(M) x NT*16(N).
// Wave w owns rows [m0, m0+16); NT accumulators cover the N tile.
// K loop in chunks of 16 (4 wmma k-steps), W chunk staged in LDS by the TDM
// (double buffered), A fragments double buffered in registers.
//
// EPI 0: in_proj : col<1024 -> x_in=silu(v) ; else -> sres=silu(v)
// EPI 1: x_projBC: col<1024 -> Bm ; else -> Cm        (W pre-offset by 16 rows)
// EPI 2: dt_proj : delta = clip(softplus(v),1e-5,0.1)
// EPI 3: out_proj: out = v
// EPI 4: x_proj-dlt (NT=1): dlt_raw[row*16+col] = v
// ---------------------------------------------------------------------------
template <int EPI, int NT, int K, int LDA, int LDW>
__global__ __launch_bounds__(256)
void gemm_wmma(const float* __restrict__ A, const float* __restrict__ W,
               const float* __restrict__ bias,
               float* __restrict__ out0, float* __restrict__ out1)
{
    __shared__ float btile[2][NT * 16 * BROW];

    const int lane = threadIdx.x & 31;
    const int wave = threadIdx.x >> 5;
    const int lr   = lane & 15;   // M (A frag) / N (B frag) index within tile
    const int hi   = lane >> 4;   // selects K pair {0,1} vs {2,3}

    const int m0 = (blockIdx.y * 8 + wave) * 16;
    const int n0 = blockIdx.x * (NT * 16);

    constexpr int NCH = K / 16;   // 16-wide K chunks

    const float* aptr = A + (size_t)(m0 + lr) * LDA + hi * 2;
    const float* wptr = W + (size_t)n0 * LDW;

    // stage chunk 0 via TDM (one wave issues; TENSORcnt tracks it)
    if (wave == 0)
        tdm_load_tile<NT * 16, LDW>(wptr, (unsigned)(uintptr_t)(void*)&btile[0][0]);

    v2f a_cur[4], a_nxt[4];
#pragma unroll
    for (int j = 0; j < 4; ++j)
        a_cur[j] = *(const v2f*)(aptr + j * 4);

    v8f acc[NT];
#pragma unroll
    for (int t = 0; t < NT; ++t) acc[t] = v8f{};

    if (wave == 0) __builtin_amdgcn_s_wait_tensorcnt(0);
    __syncthreads();

    for (int c = 0; c < NCH; ++c) {
        const int cur = c & 1;
        if (c + 1 < NCH) {
            if (wave == 0)   // prefetch next W chunk into the other LDS buffer
                tdm_load_tile<NT * 16, LDW>(wptr + (c + 1) * 16,
                    (unsigned)(uintptr_t)(void*)&btile[cur ^ 1][0]);
#pragma unroll
            for (int j = 0; j < 4; ++j)   // prefetch next A frags into regs
                a_nxt[j] = *(const v2f*)(aptr + (c + 1) * 16 + j * 4);
        }
        const float* bt = &btile[cur][0];
#pragma unroll
        for (int j = 0; j < 4; ++j) {
#pragma unroll
            for (int t = 0; t < NT; ++t) {
                const v2f b = *(const v2f*)(bt + (t * 16 + lr) * BROW + j * 4 + hi * 2);
                acc[t] = __builtin_amdgcn_wmma_f32_16x16x4_f32(
                             false, a_cur[j], false, b, (short)0, acc[t], false, false);
            }
        }
        if (c + 1 < NCH) {
            if (wave == 0) __builtin_amdgcn_s_wait_tensorcnt(0);
            __syncthreads();
#pragma unroll
            for (int j = 0; j < 4; ++j) a_cur[j] = a_nxt[j];
        }
    }

    // C/D layout: vgpr r -> row m0 + hi*8 + r ; column n0 + t*16 + lr
#pragma unroll
    for (int t = 0; t < NT; ++t) {
        const int col = n0 + t * 16 + lr;
        const float bv = bias[col];
#pragma unroll
        for (int r = 0; r < 8; ++r) {
            const int row = m0 + hi * 8 + r;
            float v = acc[t][r] + bv;
            if (EPI == 0) {
                float s = silu_f(v);
                if (col < D_MODEL) out0[(size_t)row * D_MODEL + col] = s;
                else               out1[(size_t)row * D_MODEL + (col - D_MODEL)] = s;
            } else if (EPI == 1) {
                if (col < D_MODEL) out0[(size_t)row * D_MODEL + col] = v;
                else               out1[(size_t)row * D_MODEL + (col - D_MODEL)] = v;
            } else if (EPI == 2) {
                float sp = (v > 20.0f) ? v : __logf(1.0f + __expf(v));
                sp = fminf(fmaxf(sp, 1e-5f), 0.1f);
                out0[(size_t)row * D_MODEL + col] = sp;
            } else if (EPI == 3) {
                out0[(size_t)row * D_MODEL + col] = v;
            } else {  // EPI 4: dlt_raw, row stride D_STATE
                out0[(size_t)row * D_STATE + col] = v;
            }
        }
    }
}

// ---------------------------------------------------------------------------
// Sequential selective scan: one thread per (batch, d) channel, 16-wide state
// in registers (16 independent v_exp_f32 per step). Reads silu(res) from
// `gated` and overwrites it with y*silu(res) for the out_proj GEMM.
// ---------------------------------------------------------------------------
__global__ __launch_bounds__(256)
void scan_kernel(const float* __restrict__ delta, const float* __restrict__ Bm,
                 const float* __restrict__ Cm,    const float* __restrict__ x_in,
                 float* __restrict__ gated,
                 const float* __restrict__ A_log, const float* __restrict__ D_param)
{
    const int ch = blockIdx.x * blockDim.x + threadIdx.x;   // 0..8191
    const int b  = ch >> 10;
    const int d  = ch & (D_MODEL - 1);

    float Arow[D_STATE], h[D_STATE];
#pragma unroll
    for (int s = 0; s < D_STATE; ++s) {
        float a = -__expf(A_log[d * D_STATE + s]);
        Arow[s] = fminf(fmaxf(a, -5.0f), -0.01f);
        h[s] = 0.0f;
    }
    const float Dp = D_param[d];

    size_t idx = (size_t)b * SEQ * D_MODEL + d;
    for (int t = 0; t < SEQ; ++t, idx += D_MODEL) {
        const float dt   = delta[idx];
        const float Bt   = Bm[idx];
        const float Ct   = Cm[idx];
        const float xt   = x_in[idx];
        const float sres = gated[idx];

        if (t + 1 < SEQ) {   // prefetch next timestep (4KB stride)
            __builtin_prefetch(delta + idx + D_MODEL, 0, 3);
            __builtin_prefetch(Bm    + idx + D_MODEL, 0, 3);
            __builtin_prefetch(Cm    + idx + D_MODEL, 0, 3);
            __builtin_prefetch(x_in  + idx + D_MODEL, 0, 3);
            __builtin_prefetch(gated + idx + D_MODEL, 0, 3);
        }

        const float u = Bt * dt * xt;
        float ysum = 0.0f;
#pragma unroll
        for (int s = 0; s < D_STATE; ++s) {
            float dA = __expf(Arow[s] * dt);
            dA = fminf(fmaxf(dA, 1e-6f), 1.0f);
            float hv = h[s] * dA + u;
            hv = fminf(fmaxf(hv, -100.0f), 100.0f);
            h[s] = hv;
            ysum += hv;
        }
        const float y = ysum * Ct + Dp * xt;
        gated[idx] = y * sres;
    }
}

// ---------------------------------------------------------------------------
extern "C" void kernel_launch(void* const* d_in, const int* in_sizes, int n_in,
                              void* d_out, int out_size, void* d_ws, size_t ws_size,
                              hipStream_t stream)
{
    const float* x          = (const float*)d_in[0];
    const float* in_proj_w  = (const float*)d_in[1];
    const float* in_proj_b  = (const float*)d_in[2];
    const float* x_proj_w   = (const float*)d_in[3];
    const float* x_proj_b   = (const float*)d_in[4];
    const float* dt_proj_w  = (const float*)d_in[5];
    const float* dt_proj_b  = (const float*)d_in[6];
    const float* A_log      = (const float*)d_in[7];
    const float* D_param    = (const float*)d_in[8];
    const float* out_proj_w = (const float*)d_in[9];
    const float* out_proj_b = (const float*)d_in[10];
    float* out = (float*)d_out;

    // workspace layout (floats)
    const size_t MD = (size_t)M_ROWS * D_MODEL;
    float* ws      = (float*)d_ws;
    float* x_in    = ws;                 // silu(in_proj[:, :1024])
    float* gated   = x_in  + MD;         // silu(res) -> later y*silu(res)
    float* Bm      = gated + MD;
    float* Cm      = Bm    + MD;
    float* delta   = Cm    + MD;
    float* dlt_raw = delta + MD;         // M_ROWS * D_STATE

    const dim3 blk(256);
    const int MB = M_ROWS / 128;         // 256 row-blocks

    // 1) in_proj + SiLU split: N=2048, K=1024
    gemm_wmma<0, 4, 1024, 1024, 1024><<<dim3(2048 / 64, MB), blk, 0, stream>>>(
        x, in_proj_w, in_proj_b, x_in, gated);

    // 2a) x_proj rows 16..2063 -> Bm / Cm: N=2048, K=1024
    gemm_wmma<1, 4, 1024, 1024, 1024><<<dim3(2048 / 64, MB), blk, 0, stream>>>(
        x_in, x_proj_w + (size_t)D_STATE * D_MODEL, x_proj_b + D_STATE, Bm, Cm);

    // 2b) x_proj rows 0..15 -> dlt_raw: N=16, K=1024
    gemm_wmma<4, 1, 1024, 1024, 1024><<<dim3(1, MB), blk, 0, stream>>>(
        x_in, x_proj_w, x_proj_b, dlt_raw, nullptr);

    // 3) dt_proj + softplus + clip: N=1024, K=16
    gemm_wmma<2, 4, 16, 16, 16><<<dim3(1024 / 64, MB), blk, 0, stream>>>(
        dlt_raw, dt_proj_w, dt_proj_b, delta, nullptr);

    // 4) sequential scan; writes y*silu(res) into `gated`
    scan_kernel<<<dim3((B_SZ * D_MODEL) / 256), blk, 0, stream>>>(
        delta, Bm, Cm, x_in, gated, A_log, D_param);

    // 5) out_proj: N=1024, K=1024
    gemm_wmma<3, 4, 1024, 1024, 1024><<<dim3(1024 / 64, MB), blk, 0, stream>>>(
        gated, out_proj_w, out_proj_b, out, nullptr);
}